// SingleQubitReuploadCell_87789131530342
// MI455X (gfx1250) — compile-verified
//
#include <hip/hip_runtime.h>
#include <hip/hip_bf16.h>
#include <stdint.h>

// ---------------------------------------------------------------------------
// Single-qubit data re-uploading circuit, B=1M windows x W=32 steps.
// VALU-throughput bound (~770 fp32 ops/elem vs 5.9us HBM floor), so:
//  - TDM tensor_load_to_lds DMA with LDS pad (row stride 36 floats) for the
//    x tile (confirmed lowering, round 1).
//  - Two batch elements per thread on float2 ext-vectors to get
//    v_pk_fma_f32 / v_pk_mul_f32 (2 fp32 ops per issue slot).
//  - Raw v_sin/v_cos (revolutions) with the 0.5 half-angle folded into one
//    packed multiply.
// ---------------------------------------------------------------------------

#define AS1 __attribute__((address_space(1)))
#define AS3 __attribute__((address_space(3)))

typedef unsigned int u32x4 __attribute__((ext_vector_type(4)));
typedef int          i32x4 __attribute__((ext_vector_type(4)));
typedef int          i32x8 __attribute__((ext_vector_type(8)));
typedef float        v2f   __attribute__((ext_vector_type(2)));

#ifndef __has_builtin
#define __has_builtin(x) 0
#endif

#if __has_builtin(__builtin_amdgcn_tensor_load_to_lds) && \
    __has_builtin(__builtin_amdgcn_s_wait_tensorcnt)
#define USE_TDM 1
#else
#define USE_TDM 0
#endif

#if __has_builtin(__builtin_amdgcn_sinf) && __has_builtin(__builtin_amdgcn_cosf)
#define NATIVE_TRIG 1
#else
#define NATIVE_TRIG 0
#endif

constexpr int NTHREADS = 128;  // threads per block (4 waves)
constexpr int TROWS    = 256;  // batch rows per block tile (2 per thread)
constexpr int WSTEPS   = 32;   // window length
constexpr int LSTRIDE  = 36;   // padded LDS row stride (floats): 32 data + 4 pad

__global__ __launch_bounds__(NTHREADS)
void squbit_reupload_kernel(const float* __restrict__ x,
                            const float* __restrict__ theta,
                            const float* __restrict__ w_p,
                            const float* __restrict__ b_p,
                            float* __restrict__ out, int Bn)
{
    __shared__ float  s_x[TROWS * LSTRIDE];  // padded tile: row r at r*36
    __shared__ float4 s_c[WSTEPS];           // per-step {cos_b, sin_b, cos_c, sin_c}

    const int tid  = threadIdx.x;
    const int tile = blockIdx.x * TROWS;
    int rows = Bn - tile;
    if (rows > TROWS) rows = TROWS;

    // Per-step gate constants from theta (uniform across batch): 32 lanes, once.
    if (tid < WSTEPS) {
        float hb = 0.5f * theta[2 * tid + 0];
        float hc = 0.5f * theta[2 * tid + 1];
        float sb, cb, sp, cp;
        __sincosf(hb, &sb, &cb);
        __sincosf(hc, &sp, &cp);
        s_c[tid] = make_float4(cb, sb, cp, sp);
    }

#if USE_TDM
    // One TDM DMA for the whole 256x32 fp32 tile, padded on the LDS side so
    // each row starts at a 36-float stride (reduced-bank-conflict row reads).
    if (tid < 32) {  // wave 0 of the block issues; TDM ignores EXEC
        unsigned lds_base = (unsigned)(unsigned long long)(AS3 void*)&s_x[0];
        unsigned long long ga =
            (unsigned long long)(const void*)(x + (long long)tile * WSTEPS);

        u32x4 g0;
        g0[0] = 1u;                                   // count=1, user mode
        g0[1] = lds_base;                             // lds_addr (bytes)
        g0[2] = (unsigned)(ga & 0xffffffffull);       // global_addr[31:0]
        g0[3] = (unsigned)((ga >> 32) & 0x01ffffffull) | (2u << 30); // [56:32]|type=2

        i32x8 g1;
        g1[0] = (2 << 16)      // data_size = 4 bytes
              | (1 << 20)      // pad_enable
              | (4 << 22)      // pad_interval: every 32 dwords (one row)
              | (3 << 25);     // pad_amount: 4 dwords -> row stride 36 floats
        g1[1] = (WSTEPS & 0xffff) << 16;   // tensor_dim0 = 32
        g1[2] = (rows   & 0xffff) << 16;   // tensor_dim1 = valid rows (OOB -> 0)
        g1[3] = (WSTEPS & 0xffff) << 16;   // tile_dim0 = 32
        g1[4] = TROWS & 0xffff;            // tile_dim1 = 256, tile_dim2 = 0
        g1[5] = WSTEPS;                    // tensor_dim0_stride = 32 (lo)
        g1[6] = 0;
        g1[7] = 0;

        i32x4 gz = {0, 0, 0, 0};
#if defined(__clang_major__) && (__clang_major__ >= 23)
        i32x8 gz8 = {0, 0, 0, 0, 0, 0, 0, 0};
        __builtin_amdgcn_tensor_load_to_lds(g0, g1, gz, gz, gz8, 0);
#else
        __builtin_amdgcn_tensor_load_to_lds(g0, g1, gz, gz, 0);
#endif
        __builtin_amdgcn_s_wait_tensorcnt(0);
    }
#else
    // Fallback: coalesced float4 global loads into the same padded layout.
    {
        const long long total = (long long)Bn * WSTEPS;
        #pragma unroll
        for (int j = 0; j < 16; ++j) {
            int n   = j * (NTHREADS * 4) + tid * 4;  // element index within tile
            int row = n >> 5;
            int col = n & 31;
            long long g = (long long)tile * WSTEPS + n;
            float4 v = make_float4(0.f, 0.f, 0.f, 0.f);
            if (g + 3 < total)
                v = *reinterpret_cast<const float4*>(x + g);
            *reinterpret_cast<float4*>(&s_x[row * LSTRIDE + col]) = v;
        }
    }
#endif
    __syncthreads();

    // Two elements per thread: rows tid and tid+128 of the tile.
    // State (s0, s1) = (r0+i*i0, r1+i*i1) per element, packed as float2.
    v2f r0 = {1.f, 1.f}, i0 = {0.f, 0.f}, r1 = {0.f, 0.f}, i1 = {0.f, 0.f};
    const float* xra = &s_x[tid * LSTRIDE];
    const float* xrb = &s_x[(tid + NTHREADS) * LSTRIDE];

    #pragma unroll
    for (int jq = 0; jq < 8; ++jq) {
        float4 xa = *reinterpret_cast<const float4*>(xra + 4 * jq);
        float4 xb = *reinterpret_cast<const float4*>(xrb + 4 * jq);
        float av[4] = {xa.x, xa.y, xa.z, xa.w};
        float bv[4] = {xb.x, xb.y, xb.z, xb.w};
        #pragma unroll
        for (int k = 0; k < 4; ++k) {
            const int t = 4 * jq + k;
            const float4 cc = s_c[t];            // broadcast ds_load_b128
            const v2f cb = {cc.x, cc.x};
            const v2f sb = {cc.y, cc.y};
            const v2f cp = {cc.z, cc.z};
            const v2f sp = {cc.w, cc.w};

            float sax, say, cax, cay;
#if NATIVE_TRIG
            // v_sin/v_cos take revolutions: fold 0.5 half-angle + 1/(2pi).
            const float kk = 0.07957747154594767f;  // 0.5 / (2*pi)
            v2f ang = {av[k], bv[k]};
            ang = ang * kk;                          // one v_pk_mul_f32
            sax = __builtin_amdgcn_sinf(ang.x);
            say = __builtin_amdgcn_sinf(ang.y);
            cax = __builtin_amdgcn_cosf(ang.x);
            cay = __builtin_amdgcn_cosf(ang.y);
#else
            __sincosf(0.5f * av[k], &sax, &cax);
            __sincosf(0.5f * bv[k], &say, &cay);
#endif
            const v2f sa = {sax, say};
            const v2f ca = {cax, cay};
            // RX(x_t): s0' = ca*s0 - i*sa*s1 ; s1' = -i*sa*s0 + ca*s1
            v2f ar0 = __builtin_elementwise_fma( sa, i1, ca * r0);
            v2f ai0 = __builtin_elementwise_fma(-sa, r1, ca * i0);
            v2f ar1 = __builtin_elementwise_fma( sa, i0, ca * r1);
            v2f ai1 = __builtin_elementwise_fma(-sa, r0, ca * i1);
            // RY(th0): real rotation
            v2f br0 = __builtin_elementwise_fma(-sb, ar1, cb * ar0);
            v2f bi0 = __builtin_elementwise_fma(-sb, ai1, cb * ai0);
            v2f br1 = __builtin_elementwise_fma( sb, ar0, cb * ar1);
            v2f bi1 = __builtin_elementwise_fma( sb, ai0, cb * ai1);
            // RZ(th1): s0 *= (cp - i*sp) ; s1 *= (cp + i*sp)
            r0 = __builtin_elementwise_fma( bi0, sp, br0 * cp);
            i0 = __builtin_elementwise_fma(-br0, sp, bi0 * cp);
            r1 = __builtin_elementwise_fma(-bi1, sp, br1 * cp);
            i1 = __builtin_elementwise_fma( br1, sp, bi1 * cp);
        }
    }

    const float wv = w_p[0];
    const float bv2 = b_p[0];
    v2f q = (r0 * r0 + i0 * i0) - (r1 * r1 + i1 * i1);
    const int e0 = tile + tid;
    const int e1 = tile + tid + NTHREADS;
    if (e0 < Bn) out[e0] = fmaf(q.x, wv, bv2);
    if (e1 < Bn) out[e1] = fmaf(q.y, wv, bv2);
}

extern "C" void kernel_launch(void* const* d_in, const int* in_sizes, int n_in,
                              void* d_out, int out_size, void* d_ws, size_t ws_size,
                              hipStream_t stream) {
    const float* x     = (const float*)d_in[0];
    const float* theta = (const float*)d_in[1];
    const float* w     = (const float*)d_in[2];
    const float* b     = (const float*)d_in[3];
    float* out = (float*)d_out;

    const int Bn = in_sizes[0] / WSTEPS;
    const int blocks = (Bn + TROWS - 1) / TROWS;
    squbit_reupload_kernel<<<dim3(blocks), dim3(NTHREADS), 0, stream>>>(
        x, theta, w, b, out, Bn);
}